// EctTransform_39281770889252
// MI455X (gfx1250) — compile-verified
//
#include <hip/hip_runtime.h>

// ECT transform: B=16, N=2048, D=3, T=64, R=64, SCALE=100, RADIUS=1
#define B_ 16
#define N_ 2048
#define D_ 3
#define T_ 64
#define R_ 64
#define SCALE_ 100.0f
#define NH_STRIDE 65   // padded LDS row stride (floats) to avoid bank conflicts

typedef __attribute__((ext_vector_type(2))) float v2f;
typedef __attribute__((ext_vector_type(8))) float v8f;

// ---------------------------------------------------------------------------
// Kernel 1: per (batch, N-chunk) block.
//   Stage loop: WMMA-project 128 points x 64 dirs into LDS (fp32, K=3 -> 4),
//   then each thread accumulates 16 threshold bins over the 128 points.
//   sigmoid(S*(lin-h)) = 0.5 + 0.5*tanh(S/2*(lin-h)); the 0.5 constant is
//   hoisted out of the N-sum, so the inner loop is: v_sub, v_tanh_f32, v_add
//   -- exactly one TRANS op per term (vs exp+rcp = two).
// ---------------------------------------------------------------------------
__global__ __launch_bounds__(256) void ect_accum_kernel(
    const float* __restrict__ x,     // (B, N, 3)
    const float* __restrict__ v,     // (3, T)
    float* __restrict__ partial,     // (B, nchunk, R, T)
    int nchunk)
{
    __shared__ float nh[128 * NH_STRIDE];

    const int tid   = threadIdx.x;
    const int lane  = tid & 31;          // wave32
    const int wave  = tid >> 5;          // 0..7
    const int b     = blockIdx.x / nchunk;
    const int chunk = blockIdx.x % nchunk;
    const int pointsPerBlock = N_ / nchunk;     // 128 / 512 / 2048
    const int nstages = pointsPerBlock >> 7;    // stages of 128 points
    const int pbase = chunk * pointsPerBlock;
    const bool lo16 = (lane < 16);

    // ---- B-matrix fragments for v (constant across stages): 4 t-tiles ----
    // 32-bit B 4x16 layout: lanes 0-15: vgpr0=K0, vgpr1=K1 (N=lane);
    //                       lanes 16-31: vgpr0=K2, vgpr1=K3=0 (N=lane-16).
    v2f bfrag[4];
    {
        const int nl = lane & 15;
        #pragma unroll
        for (int tt = 0; tt < 4; ++tt) {
            const int col = tt * 16 + nl;
            const float b0 = v[(lo16 ? 0 : 2) * T_ + col];
            const float b1 = v[1 * T_ + col];
            bfrag[tt].x = b0;
            bfrag[tt].y = lo16 ? b1 : 0.0f;      // K=3 zero pad
        }
    }

    // ---- per-thread threshold constants, pre-scaled for tanh(S/2 * .) ----
    const int   t  = tid & 63;           // direction handled by this thread
    const int   rg = tid >> 6;           // r-group 0..3 (16 thresholds each)
    const float H  = 0.5f * SCALE_;      // 50
    float c50[16];
    #pragma unroll
    for (int i = 0; i < 16; ++i) {
        const int r = rg * 16 + i;
        const float lin = -1.0f + 2.0f * (float)r / (float)(R_ - 1);
        c50[i] = H * lin;
    }
    float acc[16];
    #pragma unroll
    for (int i = 0; i < 16; ++i) acc[i] = 0.0f;

    const float* xb = x + (size_t)b * (N_ * D_);

    for (int s = 0; s < nstages; ++s) {
        const int p0 = pbase + s * 128;
        __syncthreads();   // previous-stage readers done before overwrite

        // ---- A fragment: 16x4 fp32, rows = 16 points of this wave ----
        // lanes 0-15: vgpr0=K0, vgpr1=K1 (M=lane); lanes 16-31: vgpr0=K2, pad.
        // Branchless: select element index / zero via cndmask, uniform loads.
        v2f afrag;
        {
            const int m = lane & 15;
            const float* xr = xb + (size_t)(p0 + wave * 16 + m) * D_;
            const float a0 = xr[lo16 ? 0 : 2];
            const float a1 = xr[1];
            afrag.x = a0;
            afrag.y = lo16 ? a1 : 0.0f;          // K=3 zero pad
        }

        // ---- 4 WMMAs: this wave's 16 points x all 64 directions ----
        #pragma unroll
        for (int tt = 0; tt < 4; ++tt) {
            v8f c = {};
            v8f d = __builtin_amdgcn_wmma_f32_16x16x4_f32(
                false, afrag, false, bfrag[tt], (short)0, c, false, false);
            // C/D layout: vgpr i -> M = i (+8 for lanes 16-31), N = lane%16
            const int rowBase = wave * 16 + (lo16 ? 0 : 8);
            const int col = tt * 16 + (lane & 15);
            #pragma unroll
            for (int i = 0; i < 8; ++i)
                nh[(rowBase + i) * NH_STRIDE + col] = d[i];
        }
        __syncthreads();

        // ---- tanh accumulation over the 128 staged points ----
        for (int p = 0; p < 128; ++p) {
            const float z = nh[p * NH_STRIDE + t] * H;   // 50*h
            #pragma unroll
            for (int i = 0; i < 16; ++i) {
                acc[i] += __builtin_amdgcn_tanhf(c50[i] - z);
            }
        }
    }

    // ---- write partial tile (B, nchunk, R, T):  0.5*(count + sum tanh) ----
    const float half_cnt = (float)pointsPerBlock;
    float* pout = partial + (size_t)(b * nchunk + chunk) * (R_ * T_);
    #pragma unroll
    for (int i = 0; i < 16; ++i)
        pout[(rg * 16 + i) * T_ + t] = 0.5f * (half_cnt + acc[i]);
}

// ---------------------------------------------------------------------------
// Kernel 2: one block per batch. Sum nchunk partials, block-max, normalize.
// Safe in-place when partial == out (all reads precede the first barrier).
// ---------------------------------------------------------------------------
__global__ __launch_bounds__(256) void ect_reduce_kernel(
    const float* __restrict__ partial, float* __restrict__ out, int nchunk)
{
    __shared__ float ect[R_ * T_];
    __shared__ float smax[256];
    const int b   = blockIdx.x;
    const int tid = threadIdx.x;
    const float* src = partial + (size_t)b * nchunk * (R_ * T_);

    float lmax = -1e30f;
    for (int idx = tid; idx < R_ * T_; idx += 256) {
        float s = 0.0f;
        for (int c = 0; c < nchunk; ++c) s += src[c * (R_ * T_) + idx];
        ect[idx] = s;
        lmax = fmaxf(lmax, s);
    }
    smax[tid] = lmax;
    __syncthreads();
    for (int off = 128; off > 0; off >>= 1) {
        if (tid < off) smax[tid] = fmaxf(smax[tid], smax[tid + off]);
        __syncthreads();
    }
    const float m = smax[0];
    float* dst = out + (size_t)b * (R_ * T_);
    for (int idx = tid; idx < R_ * T_; idx += 256)
        dst[idx] = ect[idx] / m;
}

extern "C" void kernel_launch(void* const* d_in, const int* in_sizes, int n_in,
                              void* d_out, int out_size, void* d_ws, size_t ws_size,
                              hipStream_t stream) {
    const float* x = (const float*)d_in[0];   // (16, 2048, 3) fp32
    const float* v = (const float*)d_in[1];   // (3, 64) fp32
    float* out = (float*)d_out;               // (16, 64, 64) fp32

    const size_t tileBytes = (size_t)(R_ * T_) * sizeof(float);  // 16 KB

    // Pick widest N-split the workspace allows (more blocks -> more WGPs busy).
    int nchunk;
    if      (ws_size >= (size_t)B_ * 16 * tileBytes) nchunk = 16;  // 4 MB ws
    else if (ws_size >= (size_t)B_ *  4 * tileBytes) nchunk = 4;   // 1 MB ws
    else if (ws_size >= (size_t)B_ *  1 * tileBytes) nchunk = 1;   // 256 KB ws
    else                                             nchunk = 0;   // use d_out

    float* partial = (nchunk > 0) ? (float*)d_ws : out;
    const int nc = (nchunk > 0) ? nchunk : 1;

    ect_accum_kernel<<<dim3(B_ * nc), dim3(256), 0, stream>>>(x, v, partial, nc);
    ect_reduce_kernel<<<dim3(B_), dim3(256), 0, stream>>>(partial, out, nc);
}